// Llama4MoERouter_37933151158622
// MI455X (gfx1250) — compile-verified
//
#include <hip/hip_runtime.h>

typedef __attribute__((ext_vector_type(2))) float v2f;
typedef __attribute__((ext_vector_type(4))) float v4f;
typedef __attribute__((ext_vector_type(8))) float v8f;
typedef __attribute__((ext_vector_type(4))) int   v4i;

#define TOKENS  32768
#define HIDDEN  4096
#define EXPERTS 64
#define WPB     4            // waves per block; each wave owns one 16-token M-tile
#define KCHUNK  64           // K-chunk of gate_w staged in LDS per double-buffer slot
#define NCHUNK  (HIDDEN / KCHUNK)
#define BSTRIDE 68           // padded LDS row stride (floats): conflict-free b64 fragment reads

#if __has_builtin(__builtin_amdgcn_global_load_async_to_lds_b128) && \
    __has_builtin(__builtin_amdgcn_s_wait_asynccnt)
#define USE_ASYNC 1
#else
#define USE_ASYNC 0
#endif

__device__ __forceinline__ void async_wait_all() {
#if USE_ASYNC
    __builtin_amdgcn_s_wait_asynccnt(0);
#endif
}

__global__ __launch_bounds__(WPB * 32) void
llama4_moe_router_kernel(const float* __restrict__ hs,
                         const float* __restrict__ gw,
                         float* __restrict__ out)
{
    __shared__ float bsm[2][EXPERTS][BSTRIDE];   // staged gate_w K-chunks (double buffered)
    __shared__ float red[WPB][16][EXPERTS + 1];  // epilogue tile, padded vs bank conflicts

    const int tid  = threadIdx.x;
    const int wave = tid >> 5;
    const int lane = tid & 31;
    const int tile = blockIdx.x * WPB + wave;    // 16-token M-tile
    const int m    = lane & 15;
    const int kh   = (lane >> 4) << 1;           // K sub-offset within fragment: 0 or 2

    // A stream: one hidden_states row per lane (16x4 f32 A fragment layout)
    const float* aptr = hs + (size_t)(tile * 16 + m) * HIDDEN + kh;

    // Cooperative stage of gw[0..63][k0..k0+KCHUNK) into bsm[buf]
    auto stage = [&](int buf, int k0) {
#pragma unroll
        for (int i = 0; i < (EXPERTS * KCHUNK / 4) / (WPB * 32); ++i) {  // 8 iters
            const int g   = i * (WPB * 32) + tid;       // 0..1023 (16B granules)
            const int row = g >> 4;                     // expert row
            const int seg = (g & 15) << 2;              // float offset in row
            const float* src = gw + (size_t)row * HIDDEN + k0 + seg;
#if USE_ASYNC
            __builtin_amdgcn_global_load_async_to_lds_b128(
                (v4i*)src, (v4i*)&bsm[buf][row][seg], 0, 0);
#else
            *(v4f*)&bsm[buf][row][seg] = *(const v4f*)src;
#endif
        }
    };

    v8f acc0 = {}, acc1 = {}, acc2 = {}, acc3 = {};

    stage(0, 0);
    async_wait_all();
    __syncthreads();

    for (int c = 0; c < NCHUNK; ++c) {
        const int buf = c & 1;
        if (c + 1 < NCHUNK) stage(buf ^ 1, (c + 1) * KCHUNK);   // overlap fill w/ compute

        const float* ap = aptr + c * KCHUNK;
        const int e0 = m, e1 = m + 16, e2 = m + 32, e3 = m + 48;

#pragma unroll 4
        for (int kk = 0; kk < KCHUNK; kk += 4) {
            v2f a  = *(const v2f*)(ap + kk);                    // HBM stream (b64/lane)
            v2f q0 = *(const v2f*)&bsm[buf][e0][kk + kh];       // LDS fragments (b64/lane)
            v2f q1 = *(const v2f*)&bsm[buf][e1][kk + kh];
            v2f q2 = *(const v2f*)&bsm[buf][e2][kk + kh];
            v2f q3 = *(const v2f*)&bsm[buf][e3][kk + kh];
            acc0 = __builtin_amdgcn_wmma_f32_16x16x4_f32(false, a, false, q0, (short)0, acc0, false, false);
            acc1 = __builtin_amdgcn_wmma_f32_16x16x4_f32(false, a, false, q1, (short)0, acc1, false, false);
            acc2 = __builtin_amdgcn_wmma_f32_16x16x4_f32(false, a, false, q2, (short)0, acc2, false, false);
            acc3 = __builtin_amdgcn_wmma_f32_16x16x4_f32(false, a, false, q3, (short)0, acc3, false, false);
        }
        async_wait_all();    // next chunk's LDS fill landed
        __syncthreads();     // everyone done reading current buffer
    }

    // C/D layout: VGPR r -> token row (r + 8*(lane>=16)); N = lane&15 within each 16-expert tile
    float* logits = out + (size_t)TOKENS * 4;
    const int lt_base = (lane >> 4) << 3;
    const int e = lane & 15;

#pragma unroll
    for (int r = 0; r < 8; ++r) {
        const int lt = lt_base + r;
        const size_t row = (size_t)(tile * 16 + lt) * EXPERTS;
        logits[row + e +  0] = acc0[r];
        logits[row + e + 16] = acc1[r];
        logits[row + e + 32] = acc2[r];
        logits[row + e + 48] = acc3[r];
        red[wave][lt][e +  0] = acc0[r];
        red[wave][lt][e + 16] = acc1[r];
        red[wave][lt][e + 32] = acc2[r];
        red[wave][lt][e + 48] = acc3[r];
    }

    __syncthreads();

    // Top-2: softmax denominator cancels under top-k renormalization:
    //   w1 = 1/(1+exp(l2-l1)), w2 = 1-w1  with l1 >= l2 the top-2 logits
    if (lane < 16) {
        const int token = tile * 16 + lane;
        float m1 = -3.4e38f, m2 = -3.4e38f;
        int   i1 = 0,        i2 = 0;
#pragma unroll 8
        for (int ee = 0; ee < EXPERTS; ++ee) {
            float v = red[wave][lane][ee];
            if (v > m1)      { m2 = m1; i2 = i1; m1 = v; i1 = ee; }
            else if (v > m2) { m2 = v; i2 = ee; }
        }
        const float rr = __expf(m2 - m1);
        const float w1 = 1.0f / (1.0f + rr);
        const float w2 = rr * w1;
        out[(size_t)token * 2 + 0] = w1;
        out[(size_t)token * 2 + 1] = w2;
        out[(size_t)TOKENS * 2 + (size_t)token * 2 + 0] = (float)i1;
        out[(size_t)TOKENS * 2 + (size_t)token * 2 + 1] = (float)i2;
    }
}

extern "C" void kernel_launch(void* const* d_in, const int* in_sizes, int n_in,
                              void* d_out, int out_size, void* d_ws, size_t ws_size,
                              hipStream_t stream) {
    const float* hs = (const float*)d_in[0];   // (32768, 4096) f32
    const float* gw = (const float*)d_in[1];   // (64, 4096) f32
    float* out = (float*)d_out;                // [weights | indices | logits]

    const int blocks = TOKENS / (16 * WPB);    // 512
    llama4_moe_router_kernel<<<blocks, WPB * 32, 0, stream>>>(hs, gw, out);
}